// Swin3D_45337674776737
// MI455X (gfx1250) — compile-verified
//
#include <hip/hip_runtime.h>
#include <hip/hip_bf16.h>
#include <stdint.h>

#define BB 8
#define NNODE 8000
#define NT (BB*NNODE)          // 64000
#define IN_DIM 16
#define HID 256
#define MNN 5
#define NUM_UP 2000
#define NUM_DOWN 6000
#define NEDGE (BB*NUM_DOWN*MNN)   // 240000
#define NBIN 8192

// d_out layout (flat f32, reference return order)
#define OUT_FEAT 0                      // NT*HID
#define OUT_MASK (NT*HID)               // NT
#define OUT_I    (OUT_MASK + NT)        // NEDGE
#define OUT_J    (OUT_I + NEDGE)        // NEDGE
#define OUT_SL   (OUT_J + NEDGE)        // NT*3

typedef float v2f __attribute__((ext_vector_type(2)));
typedef float v8f __attribute__((ext_vector_type(8)));

// order-preserving float<->uint encoding for atomicMax-based segment max
__device__ __forceinline__ unsigned encf(float f) {
  unsigned u = __float_as_uint(f);
  return u ^ ((u >> 31) ? 0xFFFFFFFFu : 0x80000000u);
}
__device__ __forceinline__ float decf(unsigned k) {
  unsigned u = (k & 0x80000000u) ? (k ^ 0x80000000u) : ~k;
  return __uint_as_float(u);
}
#define ENC_NEG_INF 0x007FFFFFu   // encf(-inf)

__device__ __forceinline__ int score_bin(float v) {
  int bin = (int)(v * (float)NBIN);
  return bin < 0 ? 0 : (bin > NBIN - 1 ? NBIN - 1 : bin);
}

// ---------------------------------------------------------------------------
// 1) Fuse W_h@W_att[:256] + W_coord@W_att[256:] -> W_f (16x256), and bias.
// ---------------------------------------------------------------------------
__global__ void k_fuse(const float* __restrict__ W_coord, const float* __restrict__ b_coord,
                       const float* __restrict__ W_h, const float* __restrict__ b_h,
                       const float* __restrict__ W_att, const float* __restrict__ b_att,
                       float* __restrict__ W_f, float* __restrict__ b_f) {
  int n = threadIdx.x;               // 0..255 output column
  int r = blockIdx.x;
  if (r < IN_DIM) {
    float acc = 0.f;
    for (int k = 0; k < HID; ++k) acc += W_h[r*HID + k] * W_att[k*HID + n];
    for (int k = 0; k < 3; ++k)   acc += W_coord[r*3 + k] * W_att[(HID + k)*HID + n];
    W_f[r*HID + n] = acc;
  } else {
    float acc = b_att[n];
    for (int k = 0; k < HID; ++k) acc += b_h[k] * W_att[k*HID + n];
    for (int k = 0; k < 3; ++k)   acc += b_coord[k] * W_att[(HID + k)*HID + n];
    b_f[n] = acc;
  }
}

// ---------------------------------------------------------------------------
// 2) Histogram top-NUM_UP select (O(N) instead of O(N^2) rank count).
//    Scores ~ U[0,1): 8192-bin LDS histogram, suffix scan finds the
//    threshold bin Bstar; only the ~1-element boundary bin pays an exact
//    rank loop (tie-break: larger index first, matching flip(argsort)).
// ---------------------------------------------------------------------------
__global__ void k_select(const float* __restrict__ scores,
                         int* __restrict__ mask, float* __restrict__ outMask) {
  __shared__ int hist[NBIN];         // 32 KB of 320 KB WGP LDS
  __shared__ int sBstar, sAbove;
  int b = blockIdx.x;
  const float* sc = scores + b*NNODE;
  for (int i = threadIdx.x; i < NBIN; i += blockDim.x) hist[i] = 0;
  __syncthreads();
  for (int t = threadIdx.x; t < NNODE; t += blockDim.x)
    atomicAdd(&hist[score_bin(sc[t])], 1);
  __syncthreads();
  if (threadIdx.x == 0) {
    int acc = 0, bin = NBIN - 1;
    for (; bin >= 0; --bin) {
      int c = hist[bin];
      if (acc + c >= NUM_UP) break;
      acc += c;
    }
    sBstar = bin; sAbove = acc;
  }
  __syncthreads();
  int Bstar = sBstar, above = sAbove;
  for (int t = threadIdx.x; t < NNODE; t += blockDim.x) {
    float v = sc[t];
    int bin = score_bin(v);
    int up;
    if (bin > Bstar) up = 1;
    else if (bin < Bstar) up = 0;
    else {
      int r = 0;                     // exact rank inside boundary bin only
      for (int k = 0; k < NNODE; ++k) {
        float w = sc[k];
        if (score_bin(w) == Bstar) r += (w > v) || (w == v && k > t);
      }
      up = (above + r) < NUM_UP;
    }
    mask[b*NNODE + t] = up;
    outMask[b*NNODE + t] = up ? 1.0f : 0.0f;
  }
}

// ---------------------------------------------------------------------------
// 3) Down/up node lists in ascending index order via a block-wide scan:
//    per-thread 8-element partial counts, Hillis-Steele over 1024 partials,
//    then each thread emits its 8 positions. O(N) per batch.
// ---------------------------------------------------------------------------
__global__ void k_scan(const int* __restrict__ mask,
                       int* __restrict__ down_nodes, int* __restrict__ up_nodes) {
  __shared__ unsigned char mk[NNODE];
  __shared__ int part[1024];
  int b = blockIdx.x;
  for (int t = threadIdx.x; t < NNODE; t += 1024)
    mk[t] = (unsigned char)mask[b*NNODE + t];
  __syncthreads();
  int t = threadIdx.x;
  int base = t * 8;
  int local = 0;
#pragma unroll
  for (int k = 0; k < 8; ++k) { int g = base + k; if (g < NNODE) local += mk[g]; }
  part[t] = local;
  __syncthreads();
  for (int off = 1; off < 1024; off <<= 1) {
    int v = part[t];
    int add = (t >= off) ? part[t - off] : 0;
    __syncthreads();
    part[t] = v + add;
    __syncthreads();
  }
  int upBefore = (t == 0) ? 0 : part[t - 1];
#pragma unroll
  for (int k = 0; k < 8; ++k) {
    int g = base + k;
    if (g < NNODE) {
      if (mk[g]) { up_nodes[b*NUM_UP + upBefore] = g; upBefore++; }
      else       { down_nodes[b*NUM_DOWN + (g - upBefore)] = g; }
    }
  }
}

// ---------------------------------------------------------------------------
// 4) s_l = h @ W_coord + b_coord
// ---------------------------------------------------------------------------
__global__ void k_sl(const float* __restrict__ h, const float* __restrict__ Wc,
                     const float* __restrict__ bc, float* __restrict__ outSL) {
  int t = blockIdx.x * blockDim.x + threadIdx.x;
  if (t >= NT) return;
  const float* hr = h + t*IN_DIM;
  float a0 = bc[0], a1 = bc[1], a2 = bc[2];
#pragma unroll
  for (int k = 0; k < IN_DIM; ++k) {
    float hv = hr[k];
    a0 += hv * Wc[k*3 + 0];
    a1 += hv * Wc[k*3 + 1];
    a2 += hv * Wc[k*3 + 2];
  }
  outSL[t*3 + 0] = a0; outSL[t*3 + 1] = a1; outSL[t*3 + 2] = a2;
}

// ---------------------------------------------------------------------------
// 5) feats = h @ W_f + b_f via V_WMMA_F32_16X16X4_F32. One wave owns one
//    16-wide column tile and 4 row tiles: B fragments (the fused weight)
//    are loaded once and reused across the 4 row tiles -> 16 WMMAs/wave,
//    4x fewer Wf loads. Store is branchless: down rows get feats bits, up
//    rows get the encoded -inf sentinel (in-place segmax target).
// ---------------------------------------------------------------------------
__global__ void k_feats(const float* __restrict__ h, const float* __restrict__ Wf,
                        const float* __restrict__ bf, const int* __restrict__ mask,
                        float* __restrict__ outF) {
  int lane = threadIdx.x & 31;
  int wave = threadIdx.x >> 5;
  int cn = (blockIdx.y * 8 + wave) * 16;          // column tile (16 total)
  int m  = lane & 15;
  int hi = lane >> 4;
  int n  = m;

  v2f bm[4];
#pragma unroll
  for (int s = 0; s < 4; ++s) {
    int k0 = 4*s + 2*hi;
    bm[s].x = Wf[(k0    )*HID + cn + n];
    bm[s].y = Wf[(k0 + 1)*HID + cn + n];
  }
  float bias = bf[cn + n];

#pragma unroll
  for (int rt = 0; rt < 4; ++rt) {
    int rm = (blockIdx.x * 4 + rt) * 16;          // row tile (4000 total)
    v8f c = {};
#pragma unroll
    for (int s = 0; s < 4; ++s) {
      int k0 = 4*s + 2*hi;
      v2f a;
      a.x = h[(rm + m)*IN_DIM + k0];
      a.y = h[(rm + m)*IN_DIM + k0 + 1];
      c = __builtin_amdgcn_wmma_f32_16x16x4_f32(false, a, false, bm[s],
                                                (short)0, c, false, false);
    }
#pragma unroll
    for (int r = 0; r < 8; ++r) {
      int row = rm + r + 8*hi;
      unsigned u = mask[row] ? ENC_NEG_INF : __float_as_uint(c[r] + bias);
      ((unsigned*)outF)[(size_t)row*HID + cn + n] = u;   // branchless select+store
    }
  }
}

// ---------------------------------------------------------------------------
// 6) 5-NN: each thread owns one down node, up coords + ids cached in LDS
//    (32 KB). Insertion keeps ascending d2, earlier index wins ties.
// ---------------------------------------------------------------------------
__global__ void k_nn(const float* __restrict__ sl,
                     const int* __restrict__ down_nodes, const int* __restrict__ up_nodes,
                     float* __restrict__ outI, float* __restrict__ outJ,
                     int* __restrict__ ei, int* __restrict__ ej) {
  __shared__ float ux[NUM_UP], uy[NUM_UP], uz[NUM_UP];
  __shared__ int   uid[NUM_UP];
  int b = blockIdx.x;
  for (int t = threadIdx.x; t < NUM_UP; t += blockDim.x) {
    int u = up_nodes[b*NUM_UP + t];
    uid[t] = u;
    const float* p = sl + (b*NNODE + u)*3;
    ux[t] = p[0]; uy[t] = p[1]; uz[t] = p[2];
  }
  __syncthreads();
  int d = blockIdx.y * blockDim.x + threadIdx.x;
  if (d >= NUM_DOWN) return;
  int node = down_nodes[b*NUM_DOWN + d];
  const float* p = sl + (b*NNODE + node)*3;
  float px = p[0], py = p[1], pz = p[2];
  float best[MNN]; int bidx[MNN];
#pragma unroll
  for (int m = 0; m < MNN; ++m) { best[m] = 3.4e38f; bidx[m] = 0; }
  for (int t = 0; t < NUM_UP; ++t) {
    float dx = px - ux[t], dy = py - uy[t], dz = pz - uz[t];
    float d2 = dx*dx + dy*dy + dz*dz;
    if (d2 < best[MNN-1]) {
      int m = MNN - 1;
      while (m > 0 && d2 < best[m-1]) { best[m] = best[m-1]; bidx[m] = bidx[m-1]; --m; }
      best[m] = d2; bidx[m] = t;
    }
  }
  int ig = b*NNODE + node;
#pragma unroll
  for (int m = 0; m < MNN; ++m) {
    int e  = b*(NUM_DOWN*MNN) + d*MNN + m;
    int jg = b*NNODE + uid[bidx[m]];
    ei[e] = ig; ej[e] = jg;
    outI[e] = (float)ig; outJ[e] = (float)jg;
  }
}

// ---------------------------------------------------------------------------
// 7) segment_max: 32 threads per edge, 8 dims each; order-preserving u32
//    atomicMax directly into the up rows of the features output region.
// ---------------------------------------------------------------------------
__global__ void k_segmax(const int* __restrict__ ei, const int* __restrict__ ej,
                         float* __restrict__ outF) {
  int gt = blockIdx.x * blockDim.x + threadIdx.x;
  int e = gt >> 5;
  int g = gt & 31;
  if (e >= NEDGE) return;
  int i = ei[e], j = ej[e];
  const float* src = outF + (size_t)i*HID + g*8;
  unsigned*    dst = (unsigned*)outF + (size_t)j*HID + g*8;
#pragma unroll
  for (int q = 0; q < 8; ++q) atomicMax(dst + q, encf(src[q]));
}

// ---------------------------------------------------------------------------
// 8) decode: up rows -> decoded max (0 if no incoming edge), down rows -> 0.
// ---------------------------------------------------------------------------
__global__ void k_decode(const int* __restrict__ mask, float* __restrict__ outF) {
  int row = blockIdx.x;
  int col = threadIdx.x;
  unsigned k = ((unsigned*)outF)[(size_t)row*HID + col];
  float v = 0.0f;
  if (mask[row] && k != ENC_NEG_INF) v = decf(k);
  outF[(size_t)row*HID + col] = v;
}

extern "C" void kernel_launch(void* const* d_in, const int* in_sizes, int n_in,
                              void* d_out, int out_size, void* d_ws, size_t ws_size,
                              hipStream_t stream) {
  const float* h       = (const float*)d_in[0];
  // d_in[1] = c (unused by the reference forward)
  const float* scores  = (const float*)d_in[2];
  const float* W_coord = (const float*)d_in[3];
  const float* b_coord = (const float*)d_in[4];
  const float* W_h     = (const float*)d_in[5];
  const float* b_h     = (const float*)d_in[6];
  const float* W_att   = (const float*)d_in[7];
  const float* b_att   = (const float*)d_in[8];

  float* out = (float*)d_out;
  float* outF    = out + OUT_FEAT;
  float* outMask = out + OUT_MASK;
  float* outI    = out + OUT_I;
  float* outJ    = out + OUT_J;
  float* outSL   = out + OUT_SL;

  char* ws = (char*)d_ws;
  float* W_f        = (float*)(ws);                         // 16*256 f32
  float* b_f        = (float*)(ws + 16384);                 // 256 f32
  int*   mask       = (int*)  (ws + 17408);                 // NT int
  int*   down_nodes = (int*)  (ws + 17408 + 256000);        // B*6000 int
  int*   up_nodes   = (int*)  (ws + 17408 + 256000 + 192000);
  int*   ei         = (int*)  (ws + 17408 + 256000 + 192000 + 64000);
  int*   ej         = (int*)  (ws + 17408 + 256000 + 192000 + 64000 + NEDGE*4);

  k_fuse   <<<dim3(17), dim3(256), 0, stream>>>(W_coord, b_coord, W_h, b_h, W_att, b_att, W_f, b_f);
  k_select <<<dim3(BB), dim3(1024), 0, stream>>>(scores, mask, outMask);
  k_scan   <<<dim3(BB), dim3(1024), 0, stream>>>(mask, down_nodes, up_nodes);
  k_sl     <<<dim3(NT/256), dim3(256), 0, stream>>>(h, W_coord, b_coord, outSL);
  k_feats  <<<dim3(NT/64, 2), dim3(256), 0, stream>>>(h, W_f, b_f, mask, outF);
  k_nn     <<<dim3(BB, (NUM_DOWN + 255)/256), dim3(256), 0, stream>>>(outSL, down_nodes, up_nodes, outI, outJ, ei, ej);
  k_segmax <<<dim3(NEDGE*32/256), dim3(256), 0, stream>>>(ei, ej, outF);
  k_decode <<<dim3(NT), dim3(256), 0, stream>>>(mask, outF);
}